// FunctSeparable_80659485819630
// MI455X (gfx1250) — compile-verified
//
#include <hip/hip_runtime.h>
#include <hip/hip_bf16.h>

typedef float v2f __attribute__((ext_vector_type(2)));
typedef float v8f __attribute__((ext_vector_type(8)));

#define LSZ 64
#define NBAT 64
#define NSITE 4096
#define NS_TOT (NBAT * NSITE)
#define AP 132
#define WP 132
#define Nf 4096.0f
#define Nm1 4095.0f

// ---- shared memory layout (floats) ----
constexpr int W2S_O = 0;          // 128 x 132
constexpr int W1S_O = 16896;      // 4 x 128
constexpr int B1S_O = 17408;      // 128
constexpr int B2S_O = 17536;      // 128
constexpr int W3S_O = 17664;      // 128
constexpr int WV0_O = 17808;      // per-wave slots begin
constexpr int WSLOT = 10688;      // U1,U2,A3,A4,A5 (5*2112) + 128 small
constexpr int SMEM_FLOATS = WV0_O + 4 * WSLOT;
constexpr size_t SMEM_BYTES = (size_t)SMEM_FLOATS * sizeof(float);

// ---- workspace layout (floats) ----
constexpr int WS_SUMH = 0;            // 64
constexpr int WS_SUMH2 = 64;          // 64
constexpr int WS_M = 128;             // 64
constexpr int WS_GA = 192;            // 64*3
constexpr int WS_GPP = 384;           // 64*64
constexpr int WS_SDH = 4480;          // 64*4
constexpr int WS_SHDH = 4736;         // 64*4
constexpr int WS_DM = 4992;           // 64*4
constexpr int WS_DGA = 5248;          // 64*4*3
constexpr int WS_LAPL = 6016;         // 64
constexpr int WS_GF1 = 8192;          // NS_TOT
constexpr int WS_GS = WS_GF1 + NS_TOT;
constexpr int WS_DGF1 = WS_GS + NS_TOT;       // 4*NS_TOT
constexpr int WS_DGS = WS_DGF1 + 4 * NS_TOT;  // 4*NS_TOT

// ---------------- math helpers ----------------
__device__ __forceinline__ float geluF(float x) {
  return 0.5f * x * (1.0f + erff(x * 0.70710678118654752f));
}
__device__ __forceinline__ float geluP(float x) {
  float phi = 0.3989422804014327f * expf(-0.5f * x * x);
  return 0.5f * (1.0f + erff(x * 0.70710678118654752f)) + x * phi;
}
__device__ __forceinline__ float geluPP(float x) {
  float phi = 0.3989422804014327f * expf(-0.5f * x * x);
  return phi * (2.0f - x * x);
}
__device__ __forceinline__ float lap5(const float* fld, int b, int i, int j) {
  const float* p = fld + b * NSITE;
  int im = (i + 63) & 63, ip = (i + 1) & 63, jm = (j + 63) & 63, jp = (j + 1) & 63;
  return p[im * 64 + j] + p[ip * 64 + j] + p[i * 64 + jm] + p[i * 64 + jp] - 4.0f * p[i * 64 + j];
}
__device__ __forceinline__ float zvc(const int* sg, int c, int b, int i, int j) {
  int col = ((i & 1) << 1) | (j & 1);
  if (col != c) return 0.0f;
  return 2.0f * (float)sg[((c * NBAT + b) * LSZ + i) * LSZ + j] - 1.0f;
}
__device__ __forceinline__ float lapz(const int* sg, int c, int b, int i, int j) {
  int im = (i + 63) & 63, ip = (i + 1) & 63, jm = (j + 63) & 63, jp = (j + 1) & 63;
  return zvc(sg, c, b, im, j) + zvc(sg, c, b, ip, j) + zvc(sg, c, b, i, jm) +
         zvc(sg, c, b, i, jp) - 4.0f * zvc(sg, c, b, i, j);
}
__device__ __forceinline__ float waveRed(float v) {
#pragma unroll
  for (int o = 1; o < 32; o <<= 1) v += __shfl_xor(v, o, 32);
  return v;
}

// ---------------- WMMA helpers (f32 16x16x4) ----------------
__device__ __forceinline__ v8f wmma4(v2f a, v2f b, v8f c) {
  return __builtin_amdgcn_wmma_f32_16x16x4_f32(false, a, false, b, (short)0, c, false, false);
}
__device__ __forceinline__ v8f splat8(float x) {
  v8f c;
#pragma unroll
  for (int v = 0; v < 8; ++v) c[v] = x;
  return c;
}

// ---- A-fragment register caching: whole 16x128 A-matrix = 32 v2f per lane ----
// lane m=l%16, k = kt*4 + (l/16)*2 + {0,1}
__device__ __forceinline__ void ldA_raw(v2f* a, const float* A, int lane) {
  int m = lane & 15, k0 = ((lane >> 4) << 1);
#pragma unroll
  for (int kt = 0; kt < 32; ++kt) {
    int k = kt * 4 + k0;
    v2f r; r.x = A[m * AP + k]; r.y = A[m * AP + k + 1]; a[kt] = r;
  }
}
__device__ __forceinline__ void ldA_gelu(v2f* a, const float* U1, int lane) {
  int m = lane & 15, k0 = ((lane >> 4) << 1);
#pragma unroll
  for (int kt = 0; kt < 32; ++kt) {
    int k = kt * 4 + k0;
    v2f r; r.x = geluF(U1[m * AP + k]); r.y = geluF(U1[m * AP + k + 1]); a[kt] = r;
  }
}
__device__ __forceinline__ void ldA_tan1(v2f* a, const float* U1, const float* DU1, int lane) {
  int m = lane & 15, k0 = ((lane >> 4) << 1);
#pragma unroll
  for (int kt = 0; kt < 32; ++kt) {
    int k = kt * 4 + k0;
    v2f r;
    r.x = geluP(U1[m * AP + k]) * DU1[m * AP + k];
    r.y = geluP(U1[m * AP + k + 1]) * DU1[m * AP + k + 1];
    a[kt] = r;
  }
}
__device__ __forceinline__ void ldA_gu2(v2f* a, const float* U2, const float* W3s,
                                        const float* GH, int lane) {
  int m = lane & 15, k0 = ((lane >> 4) << 1);
  float g = GH[m];
#pragma unroll
  for (int kt = 0; kt < 32; ++kt) {
    int k = kt * 4 + k0;
    v2f r;
    r.x = geluP(U2[m * AP + k]) * W3s[k] * g;
    r.y = geluP(U2[m * AP + k + 1]) * W3s[k + 1] * g;
    a[kt] = r;
  }
}
__device__ __forceinline__ void ldA_dgu2(v2f* a, const float* U2, const float* DU2,
                                         const float* W3s, const float* GH, const float* DGH,
                                         int lane) {
  int m = lane & 15, k0 = ((lane >> 4) << 1);
  float g = GH[m], dg = DGH[m];
#pragma unroll
  for (int kt = 0; kt < 32; ++kt) {
    int k = kt * 4 + k0;
    float u0 = U2[m * AP + k], u1 = U2[m * AP + k + 1];
    float d0 = DU2[m * AP + k], d1 = DU2[m * AP + k + 1];
    v2f r;
    r.x = W3s[k] * (geluPP(u0) * d0 * g + geluP(u0) * dg);
    r.y = W3s[k + 1] * (geluPP(u1) * d1 * g + geluP(u1) * dg);
    a[kt] = r;
  }
}

// B-frags: lane n=l%16, k = kt*4 + (l/16)*2 + {0,1}
__device__ __forceinline__ v2f bW2(const float* W2s, int lane, int kt, int nt) {
  int k = kt * 4 + ((lane >> 4) << 1), n = nt * 16 + (lane & 15);
  v2f r; r.x = W2s[k * WP + n]; r.y = W2s[(k + 1) * WP + n]; return r;
}
__device__ __forceinline__ v2f bW2T(const float* W2s, int lane, int kt, int nt) {
  int k = kt * 4 + ((lane >> 4) << 1), n = nt * 16 + (lane & 15);
  v2f r; r.x = W2s[n * WP + k]; r.y = W2s[n * WP + k + 1]; return r;
}
__device__ __forceinline__ v2f bW1(const float* W1s, int lane, int nt) {
  int k = ((lane >> 4) << 1), n = nt * 16 + (lane & 15);
  v2f r; r.x = W1s[k * 128 + n]; r.y = W1s[(k + 1) * 128 + n]; return r;
}
__device__ __forceinline__ v2f bW1T(const float* W1s, int lane, int kt) {
  int n = lane & 15, k = kt * 4 + ((lane >> 4) << 1);
  v2f r;
  if (n < 4) { r.x = W1s[n * 128 + k]; r.y = W1s[n * 128 + k + 1]; }
  else { r.x = 0.0f; r.y = 0.0f; }
  return r;
}
// D-frag store: row = v + 8*(l/16), col = nt*16 + l%16
__device__ __forceinline__ void stD(float* A, int lane, int nt, v8f d) {
  int n = nt * 16 + (lane & 15), mb = (lane >> 4) * 8;
#pragma unroll
  for (int v = 0; v < 8; ++v) A[(mb + v) * AP + n] = d[v];
}

// one 16-row output tile of a 128-K GEMM; BSEL: 0 -> B=W2, 1 -> B=W2^T
template <int BSEL>
__device__ __forceinline__ v8f gemmRow(const v2f* a, const float* W2s, int lane, int nt, v8f c) {
#pragma unroll
  for (int kt = 0; kt < 32; ++kt)
    c = wmma4(a[kt], BSEL ? bW2T(W2s, lane, kt, nt) : bW2(W2s, lane, kt, nt), c);
  return c;
}
__device__ __forceinline__ v8f gemmW1T(const v2f* a, const float* W1s, int lane) {
  v8f c = splat8(0.0f);
#pragma unroll
  for (int kt = 0; kt < 32; ++kt) c = wmma4(a[kt], bW1T(W1s, lane, kt), c);
  return c;
}

// ---------------- tile-level building blocks (forceinline: keep LDS addrspace) ---------------
__device__ __forceinline__ void loadWeights(float* smem, const float* W1, const float* b1,
                                            const float* W2, const float* b2, const float* W3) {
  int tid = threadIdx.x;
  for (int idx = tid; idx < 16384; idx += blockDim.x)
    smem[W2S_O + (idx >> 7) * WP + (idx & 127)] = W2[idx];
  for (int idx = tid; idx < 512; idx += blockDim.x) smem[W1S_O + idx] = W1[idx];
  for (int idx = tid; idx < 128; idx += blockDim.x) {
    smem[B1S_O + idx] = b1[idx];
    smem[B2S_O + idx] = b2[idx];
    smem[W3S_O + idx] = W3[idx];
  }
  __syncthreads();
}

__device__ __forceinline__ void fwdTile(const float* x, int s0, int lane, const float* smem,
                                        float* U1, float* U2) {
  const float* W2s = smem + W2S_O;
  const float* W1s = smem + W1S_O;
  const float* B1s = smem + B1S_O;
  const float* B2s = smem + B2S_O;
  int m = lane & 15, s = s0 + m;
  int b = s >> 12, ij = s & 4095, i = ij >> 6, j = ij & 63;
  float xc = x[s];
  v2f fa;
  if (lane < 16) { fa.x = xc; fa.y = lap5(x, b, i, j); }
  else { fa.x = xc; fa.y = xc; }
#pragma unroll
  for (int nt = 0; nt < 8; ++nt) {
    v8f c = splat8(B1s[nt * 16 + m]);
    v8f d = wmma4(fa, bW1(W1s, lane, nt), c);
    stD(U1, lane, nt, d);
  }
  __syncthreads();
  {
    v2f a[32];
    ldA_gelu(a, U1, lane);
#pragma unroll
    for (int nt = 0; nt < 8; ++nt) {
      v8f c = splat8(B2s[nt * 16 + m]);
      c = gemmRow<0>(a, W2s, lane, nt, c);
      stD(U2, lane, nt, c);
    }
  }
  __syncthreads();
}

__device__ __forceinline__ float hOfTile(const float* U2, const float* W3s, float b3, int lane) {
  int m = lane & 15, f0 = (lane >> 4) * 64;
  float p = 0.0f;
  for (int f = 0; f < 64; ++f) p += geluF(U2[m * AP + f0 + f]) * W3s[f0 + f];
  p += __shfl_xor(p, 16, 32);
  return p + b3;
}
__device__ __forceinline__ float dhOfTile(const float* U2, const float* DU2, const float* W3s,
                                          int lane) {
  int m = lane & 15, f0 = (lane >> 4) * 64;
  float p = 0.0f;
  for (int f = 0; f < 64; ++f) {
    int o = m * AP + f0 + f;
    p += geluP(U2[o]) * DU2[o] * W3s[f0 + f];
  }
  p += __shfl_xor(p, 16, 32);
  return p;
}

// ---------------- kernels ----------------
__global__ void k_zero(float* w) {
  for (int i = threadIdx.x; i < 6080; i += blockDim.x) w[i] = 0.0f;
}

__global__ __launch_bounds__(128, 1) void k_fwd_reduce(
    const float* __restrict__ x, const float* W1, const float* b1, const float* W2,
    const float* b2, const float* W3, const float* b3, float* sumh, float* sumh2) {
  extern __shared__ float smem[];
  loadWeights(smem, W1, b1, W2, b2, W3);
  int tid = threadIdx.x, lane = tid & 31, wave = tid >> 5;
  float* WSb = smem + WV0_O + wave * WSLOT;
  float* U1 = WSb; float* U2 = WSb + 2112;
  int s0 = blockIdx.x * 64 + wave * 16;
  fwdTile(x, s0, lane, smem, U1, U2);
  float h = hOfTile(U2, smem + W3S_O, b3[0], lane);
  float vh = (lane < 16) ? h : 0.0f;
  float vh2 = (lane < 16) ? h * h : 0.0f;
  vh = waveRed(vh); vh2 = waveRed(vh2);
  if (lane == 0) {
    int b = s0 >> 12;
    atomicAdd(&sumh[b], vh);
    atomicAdd(&sumh2[b], vh2);
  }
}

__global__ void k_agg(const float* sumh, const float* sumh2, const float* aW1, const float* ab1,
                      const float* aW2, float* mArr, float* gA, float* gpp) {
  int b = threadIdx.x;
  if (b >= NBAT) return;
  float sh = sumh[b], sh2 = sumh2[b];
  float m = sh / Nf, q = sh2 / Nf;
  float v = (sh2 - Nf * m * m) / Nm1;
  float g0 = 0, g1 = 0, g2 = 0;
  for (int i2 = 0; i2 < 64; ++i2) {
    float u = aW1[i2] * m + aW1[64 + i2] * v + aW1[128 + i2] * q + ab1[i2];
    float w2v = aW2[i2];
    float gg = geluP(u) * w2v;
    g0 += aW1[i2] * gg; g1 += aW1[64 + i2] * gg; g2 += aW1[128 + i2] * gg;
    gpp[b * 64 + i2] = geluPP(u) * w2v;
  }
  mArr[b] = m; gA[b * 3] = g0; gA[b * 3 + 1] = g1; gA[b * 3 + 2] = g2;
}

__global__ __launch_bounds__(128, 1) void k_grad(
    const float* __restrict__ x, const float* W1, const float* b1, const float* W2,
    const float* b2, const float* W3, const float* b3, const float* mArr, const float* gA,
    float* gsF, float* gf1F) {
  extern __shared__ float smem[];
  loadWeights(smem, W1, b1, W2, b2, W3);
  int tid = threadIdx.x, lane = tid & 31, wave = tid >> 5;
  float* WSb = smem + WV0_O + wave * WSLOT;
  float* U1 = WSb; float* U2 = WSb + 2112; float* A3 = WSb + 4224;
  float* SS = WSb + 10560; float* GH = SS; float* DGF = SS + 64;
  const float* W2s = smem + W2S_O; const float* W1s = smem + W1S_O; const float* W3s = smem + W3S_O;
  int s0 = blockIdx.x * 64 + wave * 16;
  int b = s0 >> 12;
  fwdTile(x, s0, lane, smem, U1, U2);
  float h = hOfTile(U2, W3s, b3[0], lane);
  float mb = mArr[b], g0 = gA[b * 3], g1 = gA[b * 3 + 1], g2 = gA[b * 3 + 2];
  float gh = g0 / Nf + g1 * 2.0f * (h - mb) / Nm1 + g2 * 2.0f * h / Nf;
  if (lane < 16) GH[lane] = gh;
  __syncthreads();
  v2f a[32];
  // gh1 = gu2 @ W2^T ; combine to gu1, store into A3
  ldA_gu2(a, U2, W3s, GH, lane);
#pragma unroll
  for (int nt = 0; nt < 8; ++nt) {
    v8f c = gemmRow<1>(a, W2s, lane, nt, splat8(0.0f));
    int n = nt * 16 + (lane & 15), mb8 = (lane >> 4) * 8;
#pragma unroll
    for (int v = 0; v < 8; ++v) {
      int row = mb8 + v;
      A3[row * AP + n] = geluP(U1[row * AP + n]) * c[v];
    }
  }
  __syncthreads();
  // gf = gu1 @ W1^T
  {
    ldA_raw(a, A3, lane);
    v8f c = gemmW1T(a, W1s, lane);
    int col = lane & 15, mb8 = (lane >> 4) * 8;
    if (col < 4) {
#pragma unroll
      for (int v = 0; v < 8; ++v) DGF[(mb8 + v) * 4 + col] = c[v];
    }
  }
  __syncthreads();
  if (lane < 16) {
    int mm = lane;
    gsF[s0 + mm] = DGF[mm * 4 + 0] + DGF[mm * 4 + 2] + DGF[mm * 4 + 3];
    gf1F[s0 + mm] = DGF[mm * 4 + 1];
  }
}

__global__ void k_grad_out(const float* gsF, const float* gf1F, float* out) {
  int s = blockIdx.x * blockDim.x + threadIdx.x;
  if (s >= NS_TOT) return;
  int b = s >> 12, ij = s & 4095, i = ij >> 6, j = ij & 63;
  out[b * 4097 + ij] = gsF[s] + lap5(gf1F, b, i, j);
}

__global__ __launch_bounds__(128, 1) void k_tfwd(
    const float* __restrict__ x, const int* __restrict__ signs, const float* W1, const float* b1,
    const float* W2, const float* b2, const float* W3, const float* b3, float* sdh, float* shdh) {
  extern __shared__ float smem[];
  loadWeights(smem, W1, b1, W2, b2, W3);
  int tid = threadIdx.x, lane = tid & 31, wave = tid >> 5;
  float* WSb = smem + WV0_O + wave * WSLOT;
  float* U1 = WSb; float* U2 = WSb + 2112; float* A4 = WSb + 6336; float* A5 = WSb + 8448;
  const float* W2s = smem + W2S_O; const float* W1s = smem + W1S_O; const float* W3s = smem + W3S_O;
  int s0 = blockIdx.x * 64 + wave * 16;
  int b = s0 >> 12;
  fwdTile(x, s0, lane, smem, U1, U2);
  float h = hOfTile(U2, W3s, b3[0], lane);
  int m = lane & 15, s = s0 + m, ij = s & 4095, i = ij >> 6, j = ij & 63;
  v2f a[32];
  for (int c = 0; c < 4; ++c) {
    float zc = zvc(signs, c, b, i, j);
    v2f dfa;
    if (lane < 16) { dfa.x = zc; dfa.y = lapz(signs, c, b, i, j); }
    else { dfa.x = zc; dfa.y = zc; }
#pragma unroll
    for (int nt = 0; nt < 8; ++nt) stD(A4, lane, nt, wmma4(dfa, bW1(W1s, lane, nt), splat8(0.0f)));
    __syncthreads();
    ldA_tan1(a, U1, A4, lane);
#pragma unroll
    for (int nt = 0; nt < 8; ++nt) {
      v8f cc = gemmRow<0>(a, W2s, lane, nt, splat8(0.0f));
      stD(A5, lane, nt, cc);
    }
    __syncthreads();
    float dh = dhOfTile(U2, A5, W3s, lane);
    float vd = (lane < 16) ? dh : 0.0f;
    float vhd = (lane < 16) ? h * dh : 0.0f;
    vd = waveRed(vd); vhd = waveRed(vhd);
    if (lane == 0) {
      atomicAdd(&sdh[b * 4 + c], vd);
      atomicAdd(&shdh[b * 4 + c], vhd);
    }
    __syncthreads();
  }
}

__global__ void k_tagg(const float* sdh, const float* shdh, const float* mArr, const float* gpp,
                       const float* aW1, float* dmA, float* dgA) {
  int t = threadIdx.x;
  if (t >= 256) return;
  int b = t >> 2;
  float sd = sdh[t], shd = shdh[t], m = mArr[b];
  float dm = sd / Nf;
  float dv = 2.0f * (shd - Nf * m * dm) / Nm1;
  float dq = 2.0f * shd / Nf;
  float d0 = 0, d1 = 0, d2 = 0;
  for (int i2 = 0; i2 < 64; ++i2) {
    float du = aW1[i2] * dm + aW1[64 + i2] * dv + aW1[128 + i2] * dq;
    float dg = gpp[b * 64 + i2] * du;
    d0 += aW1[i2] * dg; d1 += aW1[64 + i2] * dg; d2 += aW1[128 + i2] * dg;
  }
  dmA[t] = dm; dgA[t * 3] = d0; dgA[t * 3 + 1] = d1; dgA[t * 3 + 2] = d2;
}

__global__ __launch_bounds__(128, 1) void k_hvp(
    const float* __restrict__ x, const int* __restrict__ signs, const float* W1, const float* b1,
    const float* W2, const float* b2, const float* W3, const float* b3, const float* mArr,
    const float* gA, const float* dmA, const float* dgAa, float* dgsF, float* dgf1F) {
  extern __shared__ float smem[];
  loadWeights(smem, W1, b1, W2, b2, W3);
  int tid = threadIdx.x, lane = tid & 31, wave = tid >> 5;
  float* WSb = smem + WV0_O + wave * WSLOT;
  float* U1 = WSb; float* U2 = WSb + 2112; float* A3 = WSb + 4224;
  float* A4 = WSb + 6336; float* A5 = WSb + 8448;
  float* SS = WSb + 10560; float* GH = SS; float* DGH = SS + 16; float* DGF = SS + 64;
  const float* W2s = smem + W2S_O; const float* W1s = smem + W1S_O; const float* W3s = smem + W3S_O;
  int s0 = blockIdx.x * 64 + wave * 16;
  int b = s0 >> 12;
  fwdTile(x, s0, lane, smem, U1, U2);
  float h = hOfTile(U2, W3s, b3[0], lane);
  float mb = mArr[b], g1 = gA[b * 3 + 1], g2 = gA[b * 3 + 2];
  float gh = gA[b * 3] / Nf + g1 * 2.0f * (h - mb) / Nm1 + g2 * 2.0f * h / Nf;
  if (lane < 16) GH[lane] = gh;
  __syncthreads();
  v2f a[32];
  // primal gh1 = gu2 @ W2^T stored raw in A3
  ldA_gu2(a, U2, W3s, GH, lane);
#pragma unroll
  for (int nt = 0; nt < 8; ++nt) {
    v8f c = gemmRow<1>(a, W2s, lane, nt, splat8(0.0f));
    stD(A3, lane, nt, c);
  }
  __syncthreads();
  int m = lane & 15, s = s0 + m, ij = s & 4095, i = ij >> 6, j = ij & 63;
  for (int c = 0; c < 4; ++c) {
    // tangent forward: du1 -> A4, du2 -> A5
    float zc = zvc(signs, c, b, i, j);
    v2f dfa;
    if (lane < 16) { dfa.x = zc; dfa.y = lapz(signs, c, b, i, j); }
    else { dfa.x = zc; dfa.y = zc; }
#pragma unroll
    for (int nt = 0; nt < 8; ++nt) stD(A4, lane, nt, wmma4(dfa, bW1(W1s, lane, nt), splat8(0.0f)));
    __syncthreads();
    ldA_tan1(a, U1, A4, lane);
#pragma unroll
    for (int nt = 0; nt < 8; ++nt) {
      v8f cc = gemmRow<0>(a, W2s, lane, nt, splat8(0.0f));
      stD(A5, lane, nt, cc);
    }
    __syncthreads();
    float dh = dhOfTile(U2, A5, W3s, lane);
    float dmv = dmA[b * 4 + c];
    const float* dg = dgAa + (b * 4 + c) * 3;
    float dgh = dg[0] / Nf + dg[1] * 2.0f * (h - mb) / Nm1 + g1 * 2.0f * (dh - dmv) / Nm1 +
                dg[2] * 2.0f * h / Nf + g2 * 2.0f * dh / Nf;
    if (lane < 16) DGH[lane] = dgh;
    __syncthreads();
    // d_gh1 = d_gu2 @ W2^T ; combine to d_gu1 (overwrite A4)
    ldA_dgu2(a, U2, A5, W3s, GH, DGH, lane);
#pragma unroll
    for (int nt = 0; nt < 8; ++nt) {
      v8f cc = gemmRow<1>(a, W2s, lane, nt, splat8(0.0f));
      int n = nt * 16 + (lane & 15), mb8 = (lane >> 4) * 8;
#pragma unroll
      for (int v = 0; v < 8; ++v) {
        int row = mb8 + v, o = row * AP + n;
        float u1 = U1[o];
        A4[o] = geluPP(u1) * A4[o] * A3[o] + geluP(u1) * cc[v];
      }
    }
    __syncthreads();
    // d_gf = d_gu1 @ W1^T
    {
      ldA_raw(a, A4, lane);
      v8f cc = gemmW1T(a, W1s, lane);
      int col = lane & 15, mb8 = (lane >> 4) * 8;
      if (col < 4) {
#pragma unroll
        for (int v = 0; v < 8; ++v) DGF[(mb8 + v) * 4 + col] = cc[v];
      }
    }
    __syncthreads();
    if (lane < 16) {
      dgsF[c * NS_TOT + s0 + lane] = DGF[lane * 4 + 0] + DGF[lane * 4 + 2] + DGF[lane * 4 + 3];
      dgf1F[c * NS_TOT + s0 + lane] = DGF[lane * 4 + 1];
    }
    __syncthreads();
  }
}

__global__ void k_lapl(const float* dgsF, const float* dgf1F, const int* signs, float* lapl) {
  __shared__ float red[256];
  int tid = threadIdx.x;
  int s = blockIdx.x * 256 + tid;
  int b = s >> 12, ij = s & 4095, i = ij >> 6, j = ij & 63;
  int c = ((i & 1) << 1) | (j & 1);
  float zc = 2.0f * (float)signs[((c * NBAT + b) * LSZ + i) * LSZ + j] - 1.0f;
  float hv = dgsF[c * NS_TOT + s] + lap5(dgf1F + c * NS_TOT, b, i, j);
  red[tid] = hv * zc;
  __syncthreads();
  for (int o = 128; o > 0; o >>= 1) {
    if (tid < o) red[tid] += red[tid + o];
    __syncthreads();
  }
  if (tid == 0) atomicAdd(&lapl[b], red[0]);
}

__global__ void k_finish(const float* lapl, float* out) {
  int b = threadIdx.x;
  if (b < NBAT) out[b * 4097 + 4096] = lapl[b];
}

// ---------------- host launch ----------------
extern "C" void kernel_launch(void* const* d_in, const int* in_sizes, int n_in,
                              void* d_out, int out_size, void* d_ws, size_t ws_size,
                              hipStream_t stream) {
  const float* x = (const float*)d_in[0];
  const int* signs = (const int*)d_in[1];
  const float* W1 = (const float*)d_in[2];
  const float* b1 = (const float*)d_in[3];
  const float* W2 = (const float*)d_in[4];
  const float* b2 = (const float*)d_in[5];
  const float* W3 = (const float*)d_in[6];
  const float* b3 = (const float*)d_in[7];
  const float* aW1 = (const float*)d_in[8];
  const float* ab1 = (const float*)d_in[9];
  const float* aW2 = (const float*)d_in[10];
  float* out = (float*)d_out;
  float* w = (float*)d_ws;

  float* sumh = w + WS_SUMH;
  float* sumh2 = w + WS_SUMH2;
  float* mArr = w + WS_M;
  float* gA = w + WS_GA;
  float* gpp = w + WS_GPP;
  float* sdh = w + WS_SDH;
  float* shdh = w + WS_SHDH;
  float* dmA = w + WS_DM;
  float* dgA = w + WS_DGA;
  float* lapl = w + WS_LAPL;
  float* gf1F = w + WS_GF1;
  float* gsF = w + WS_GS;
  float* dgf1F = w + WS_DGF1;
  float* dgsF = w + WS_DGS;

  const int HG = NS_TOT / 64;  // 4096 workgroups of 4 waves x 16-site tiles

  k_zero<<<1, 256, 0, stream>>>(w);
  k_fwd_reduce<<<HG, 128, SMEM_BYTES, stream>>>(x, W1, b1, W2, b2, W3, b3, sumh, sumh2);
  k_agg<<<1, 64, 0, stream>>>(sumh, sumh2, aW1, ab1, aW2, mArr, gA, gpp);
  k_grad<<<HG, 128, SMEM_BYTES, stream>>>(x, W1, b1, W2, b2, W3, b3, mArr, gA, gsF, gf1F);
  k_grad_out<<<NS_TOT / 256, 256, 0, stream>>>(gsF, gf1F, out);
  k_tfwd<<<HG, 128, SMEM_BYTES, stream>>>(x, signs, W1, b1, W2, b2, W3, b3, sdh, shdh);
  k_tagg<<<1, 256, 0, stream>>>(sdh, shdh, mArr, gpp, aW1, dmA, dgA);
  k_hvp<<<HG, 128, SMEM_BYTES, stream>>>(x, signs, W1, b1, W2, b2, W3, b3, mArr, gA, dmA, dgA,
                                         dgsF, dgf1F);
  k_lapl<<<NS_TOT / 256, 256, 0, stream>>>(dgsF, dgf1F, signs, lapl);
  k_finish<<<1, 64, 0, stream>>>(lapl, out);
}